// RegressionLoss_9612136808649
// MI455X (gfx1250) — compile-verified
//
#include <hip/hip_runtime.h>
#include <cstdint>
#include <cstddef>

#define N_ANN   32
#define A_TOTAL 258048
#define BLK     256
#define NBLK    1008      /* A_TOTAL / BLK exactly */

typedef float v2f __attribute__((ext_vector_type(2)));
typedef float v8f __attribute__((ext_vector_type(8)));

#if defined(__HIP_DEVICE_COMPILE__) && defined(__gfx1250__)
#if __has_builtin(__builtin_amdgcn_global_load_async_to_lds_b32)
#define HAVE_ASYNC_LDS 1
#endif
#if __has_builtin(__builtin_amdgcn_wmma_f32_16x16x4_f32)
#define HAVE_WMMA_F32 1
#endif
#endif

// ---------------------------------------------------------------------------
// Kernel 1: stable sort of 32 annotations by length (rank sort), per sample.
// ---------------------------------------------------------------------------
__global__ void rl_prep(const float* __restrict__ ann,   // (B, 32, 3)
                        float* __restrict__ sS,          // (B, 32)
                        float* __restrict__ sE) {        // (B, 32)
    const int b = blockIdx.x;
    const int t = threadIdx.x;
    __shared__ float ln_[N_ANN];
    const float* ap = ann + (size_t)(b * N_ANN + t) * 3;
    const float s = ap[0];
    const float e = ap[1];
    ln_[t] = e - s;
    __syncthreads();
    const float myLen = ln_[t];
    int rank = 0;
#pragma unroll
    for (int u = 0; u < N_ANN; ++u) {
        const float lu = ln_[u];
        rank += (lu < myLen || (lu == myLen && u < t)) ? 1 : 0;
    }
    sS[b * N_ANN + rank] = s;
    sE[b * N_ANN + rank] = e;
}

// ---------------------------------------------------------------------------
// Kernel 2: per-anchor positive matching + GIoU loss, block partial sums.
// ---------------------------------------------------------------------------
__global__ void rl_main(const float* __restrict__ reg,   // (B, A, 2)
                        const float* __restrict__ sS,
                        const float* __restrict__ sE,
                        float* __restrict__ pLoss,       // (B, NBLK)
                        float* __restrict__ pPos) {      // (B, NBLK)
    constexpr double RATE = 22050.0 / 256.0;
    const float SZ0 = (float)(2.23147392 * RATE);
    const float SZ1 = (float)(2.62519274 * RATE);
    const float SZ2 = (float)(3.74199546 * RATE);
    const float SZ3 = (float)(5.78800454 * RATE);
    const float SZ4 = (float)(8.02371882 * RATE);
    const float INF = __builtin_inff();
    const float EPS = 1e-7f;

    const int b   = blockIdx.y;
    const int blk = blockIdx.x;
    const int tid = threadIdx.x;
    const int a   = blk * BLK + tid;   // always < A_TOTAL (NBLK*BLK == A_TOTAL)

    __shared__ float sAnn[2 * N_ANN];  // [0..31]=sortedS, [32..63]=sortedE

#if defined(HAVE_ASYNC_LDS)
    if (tid < 2 * N_ANN) {
        const float* g = (tid < N_ANN) ? (sS + b * N_ANN + tid)
                                       : (sE + b * N_ANN + (tid - N_ANN));
        unsigned long long gi = (unsigned long long)(uintptr_t)(const void*)g;
        __attribute__((address_space(1))) int* gp =
            (__attribute__((address_space(1))) int*)gi;
        unsigned int li = (unsigned int)(uintptr_t)(void*)&sAnn[tid];
        __attribute__((address_space(3))) int* lp =
            (__attribute__((address_space(3))) int*)li;
        __builtin_amdgcn_global_load_async_to_lds_b32(gp, lp, 0, 0);
    }
#if __has_builtin(__builtin_amdgcn_s_wait_asynccnt)
    __builtin_amdgcn_s_wait_asynccnt(0);
#else
    asm volatile("s_wait_asynccnt 0" ::: "memory");
#endif
    __syncthreads();
#else
    if (tid < 2 * N_ANN) {
        sAnn[tid] = (tid < N_ANN) ? sS[b * N_ANN + tid]
                                  : sE[b * N_ANN + (tid - N_ANN)];
    }
    __syncthreads();
#endif

    // Level bracket for this anchor (concatenated pyramid order).
    float scale, lower, upper;
    int off;
    if (a < 131072)      { scale = 1.f;  off = 0;      lower = 0.f; upper = SZ0; }
    else if (a < 196608) { scale = 2.f;  off = 131072; lower = SZ0; upper = SZ1; }
    else if (a < 229376) { scale = 4.f;  off = 196608; lower = SZ1; upper = SZ2; }
    else if (a < 245760) { scale = 8.f;  off = 229376; lower = SZ2; upper = SZ3; }
    else if (a < 253952) { scale = 16.f; off = 245760; lower = SZ3; upper = SZ4; }
    else                 { scale = 32.f; off = 253952; lower = SZ4; upper = INF; }
    const float inv = 1.0f / scale;                       // exact power of two
    const float pt  = ((float)(a - off) + 0.5f) * scale;  // == anchors_l{i}[k]

    // First (shortest) annotation whose bracket matches; default index 0.
    float chosenL = pt - sAnn[0];
    float chosenR = sAnn[N_ANN] - pt;
    bool  found   = false;
#pragma unroll
    for (int j = 0; j < N_ANN; ++j) {
        const float l = pt - sAnn[j];
        const float r = sAnn[N_ANN + j] - pt;
        const float m = fmaxf(l, r);
        const bool cand = (l >= 0.f) && (r >= 0.f) && (m >= lower) && (m < upper);
        const bool take = cand && !found;
        chosenL = take ? l : chosenL;
        chosenR = take ? r : chosenR;
        found   = found || cand;
    }

    const float posf = found ? 1.0f : 0.0f;
    const float nl = chosenL * inv;
    const float nr = chosenR * inv;
    const float bs = pt - nl;
    const float be = pt + nr;
    const float2 rr = ((const float2*)reg)[(size_t)b * A_TOTAL + a];
    const float ra = rr.x, rb = rr.y;

    const float inter = fmaxf(fminf(be, rb) - fmaxf(bs, ra), 0.0f);
    const float uni   = (be - bs) + (rb - ra) - inter;
    const float iou   = inter / (uni + EPS);
    const float enc   = fmaxf(be, rb) - fminf(bs, ra);
    float giou = iou - (enc - uni) / (enc + EPS);
    giou = fminf(fmaxf(giou, -1.0f), 1.0f);
    const float loss = (1.0f - giou) * posf;

    // Deterministic block tree reduction.
    __shared__ float rL[BLK];
    __shared__ float rP[BLK];
    rL[tid] = loss;
    rP[tid] = posf;
    __syncthreads();
    for (int s = BLK / 2; s > 0; s >>= 1) {
        if (tid < s) { rL[tid] += rL[tid + s]; rP[tid] += rP[tid + s]; }
        __syncthreads();
    }
    if (tid == 0) {
        pLoss[b * NBLK + blk] = rL[0];
        pPos [b * NBLK + blk] = rP[0];
    }
}

// ---------------------------------------------------------------------------
// Kernel 3: one wave per sample. WMMA f32 16x16x4 with A==ones reduces 64
// partials per issue; all 16 D rows are identical column-sums, so
// sum(all D entries)/16 is the exact total regardless of VGPR layout.
// ---------------------------------------------------------------------------
__global__ void rl_final(const float* __restrict__ pLoss,
                         const float* __restrict__ pPos,
                         float* __restrict__ out) {
    const int b    = blockIdx.x;
    const int lane = threadIdx.x;
    const float* PL = pLoss + b * NBLK;
    const float* PP = pPos  + b * NBLK;

    float sL, sP;
#if defined(HAVE_WMMA_F32)
    v2f av; av.x = 1.0f; av.y = 1.0f;          // A = ones(16x4)
    v8f cL = {0.f,0.f,0.f,0.f,0.f,0.f,0.f,0.f};
    v8f cP = {0.f,0.f,0.f,0.f,0.f,0.f,0.f,0.f};
    for (int base = 0; base < NBLK; base += 64) {
        const int i0 = base + lane;
        const int i1 = base + 32 + lane;
        v2f bl; bl.x = 0.f; bl.y = 0.f;
        v2f bp; bp.x = 0.f; bp.y = 0.f;
        if (i0 < NBLK) { bl.x = PL[i0]; bp.x = PP[i0]; }
        if (i1 < NBLK) { bl.y = PL[i1]; bp.y = PP[i1]; }
        // EXEC is all-ones here (wave32 block of 32, reconverged).
        cL = __builtin_amdgcn_wmma_f32_16x16x4_f32(false, av, false, bl,
                                                   (short)0, cL, false, false);
        cP = __builtin_amdgcn_wmma_f32_16x16x4_f32(false, av, false, bp,
                                                   (short)0, cP, false, false);
    }
    sL = 0.f; sP = 0.f;
#pragma unroll
    for (int k = 0; k < 8; ++k) { sL += cL[k]; sP += cP[k]; }
#else
    sL = 0.f; sP = 0.f;
    for (int i = lane; i < NBLK; i += 32) { sL += PL[i]; sP += PP[i]; }
#endif
#pragma unroll
    for (int m = 16; m >= 1; m >>= 1) {
        sL += __shfl_xor(sL, m, 32);
        sP += __shfl_xor(sP, m, 32);
    }
#if defined(HAVE_WMMA_F32)
    sL *= (1.0f / 16.0f);   // 16 identical D rows -> exact
    sP *= (1.0f / 16.0f);
#endif
    if (lane == 0) out[b] = sL / fmaxf(sP, 1.0f);   // WEIGHT == 1.0
}

// ---------------------------------------------------------------------------
extern "C" void kernel_launch(void* const* d_in, const int* in_sizes, int n_in,
                              void* d_out, int out_size, void* d_ws, size_t ws_size,
                              hipStream_t stream) {
    (void)n_in; (void)out_size; (void)ws_size;
    const float* reg = (const float*)d_in[0];   // (B, A, 2)
    const float* ann = (const float*)d_in[1];   // (B, 32, 3)
    // d_in[2] = class_id (unused); d_in[3..8] = anchors (recomputed exactly)

    const int B = in_sizes[1] / (N_ANN * 3);

    float* ws    = (float*)d_ws;
    float* sS    = ws;                          // B*32
    float* sE    = sS + (size_t)B * N_ANN;      // B*32
    float* pLoss = sE + (size_t)B * N_ANN;      // B*NBLK
    float* pPos  = pLoss + (size_t)B * NBLK;    // B*NBLK

    rl_prep<<<dim3(B), dim3(N_ANN), 0, stream>>>(ann, sS, sE);
    rl_main<<<dim3(NBLK, B), dim3(BLK), 0, stream>>>(reg, sS, sE, pLoss, pPos);
    rl_final<<<dim3(B), dim3(32), 0, stream>>>(pLoss, pPos, (float*)d_out);
}